// Flow_773094113598
// MI455X (gfx1250) — compile-verified
//
#include <hip/hip_runtime.h>
#include <math.h>

namespace {
constexpr int   kLayers = 10;
constexpr float kLog2Pi = 1.8378770664093453f;
constexpr float kEps    = 1.0e-4f;
constexpr int   kBlock  = 256;          // 8 waves (wave32)
constexpr int   kRowF   = 60;           // affine_info floats per row
constexpr int   kRowB   = kRowF * 4;    // 240 bytes per row
} // namespace

// sigmoid(r + 2) + eps
__device__ __forceinline__ float fast_sigmoid_p2(float r) {
#if __has_builtin(__builtin_amdgcn_tanhf)
  // gfx1250 native v_tanh_f32: sigmoid(y) = 0.5*tanh(y/2) + 0.5 (exact identity)
  float h = fmaf(0.5f, r, 1.0f);                       // (r+2)/2
  return fmaf(0.5f, __builtin_amdgcn_tanhf(h), 0.5f + kEps);
#else
  float e = __expf(-(r + 2.0f));
  return __builtin_amdgcn_rcpf(1.0f + e) + kEps;
#endif
}

// ---- Prepass: sum of log|det(W_i)| (+ the Gaussian normalization constant)
//      is dispatch-wide; compute once on one lane instead of in 2M threads.
//      Lowers to gfx1250 scalar-float SALU; perf-irrelevant. ----
__global__ void logdet_prepass(const float* __restrict__ Ws, float* __restrict__ out) {
  if (threadIdx.x == 0 && blockIdx.x == 0) {
    float c = -1.5f * kLog2Pi;   // sum_n -0.5*log(2*pi), n = 0..2
#pragma clang loop unroll(disable)
    for (int i = 0; i <= kLayers; ++i) {
      const float* w = Ws + i * 9;
      float m0 = fmaf(w[4], w[8], -w[5] * w[7]);
      float m1 = fmaf(w[3], w[8], -w[5] * w[6]);
      float m2 = fmaf(w[3], w[7], -w[4] * w[6]);
      float det = fmaf(w[0], m0, fmaf(-w[1], m1, w[2] * m2));
      c += __logf(fabsf(det));
    }
    out[0] = c;
  }
}

__global__ __launch_bounds__(kBlock)
void flow_fwd_kernel(const float* __restrict__ x,
                     const float* __restrict__ aff,
                     const float* __restrict__ Ws,     // (11,3,3) row-major
                     const float* __restrict__ bs,     // (11,3)
                     const float* __restrict__ cdetIn, // [0] = sum log|det W| - 1.5*log2pi
                     float* __restrict__ outZ,         // (N,3)
                     float* __restrict__ outT,         // (N,)
                     int n)
{
  __shared__ float saff[kBlock * kRowF];   // 61440 B
  __shared__ float sx[kBlock * 3];         //  3072 B  (total 64512 B -> 5 blocks/WGP)

  const int tid  = threadIdx.x;
  const int wave = tid >> 5;
  const int lane = tid & 31;
  const long long rowBase = (long long)blockIdx.x * kBlock;
  const long long row     = rowBase + tid;

  const bool fullBlock = (rowBase + kBlock) <= (long long)n;

  if (fullBlock) {
    // ---- CDNA5 async global->LDS staging, ASYNCcnt-tracked, wave-self-
    //      contained (each wave loads exactly its own 32 rows -> no barrier).
    // affine_info: 32 rows * 240 B = 7680 B = 15 coalesced 16B chunks / lane
    const long long waveRow = rowBase + (long long)wave * 32;
    {
      unsigned long long gwave =
          (unsigned long long)(uintptr_t)aff + (unsigned long long)waveRow * (unsigned)kRowB;
      unsigned ldsWave =
          (unsigned)(uintptr_t)(&saff[0]) + (unsigned)wave * (32u * (unsigned)kRowB);
#pragma unroll
      for (int i = 0; i < 15; ++i) {
        unsigned off = (unsigned)lane * 16u + (unsigned)i * 512u;
        unsigned ldsAddr = ldsWave + off;
        unsigned long long gAddr = gwave + off;
        asm volatile("global_load_async_to_lds_b128 %0, %1, off th:TH_LOAD_NT"
                     :: "v"(ldsAddr), "v"(gAddr)
                     : "memory");
      }
    }
    // x: 32 rows * 12 B = 384 B = 24 chunks; lanes 0..23 issue one chunk each
    if (lane < 24) {
      unsigned long long gAddr =
          (unsigned long long)(uintptr_t)x + (unsigned long long)waveRow * 12u +
          (unsigned)lane * 16u;
      unsigned ldsAddr =
          (unsigned)(uintptr_t)(&sx[0]) + (unsigned)wave * 384u + (unsigned)lane * 16u;
      asm volatile("global_load_async_to_lds_b128 %0, %1, off th:TH_LOAD_NT"
                   :: "v"(ldsAddr), "v"(gAddr)
                   : "memory");
    }
  } else if (row < n) {
    // tail fallback (never taken when N % 256 == 0): direct copy into LDS
#pragma unroll 4
    for (int k = 0; k < kRowF; ++k) saff[tid * kRowF + k] = aff[row * kRowF + k];
    sx[tid * 3 + 0] = x[row * 3 + 0];
    sx[tid * 3 + 1] = x[row * 3 + 1];
    sx[tid * 3 + 2] = x[row * 3 + 2];
  }

  // uniform scalar load overlaps with in-flight async copies
  const float cdet = cdetIn[0];

  asm volatile("s_wait_asynccnt 0x0" ::: "memory");

  if (row >= n) return;

  float z0 = sx[tid * 3 + 0];
  float z1 = sx[tid * 3 + 1];
  float z2 = sx[tid * 3 + 2];

  const float* a = &saff[tid * kRowF];
  float total = cdet;

#pragma unroll
  for (int i = 0; i < kLayers; ++i) {
    const float* w = Ws + i * 9;
    const float* b = bs + i * 3;
    // z = W z + b   (z @ W^T, W row-major)
    float t0 = fmaf(w[0], z0, fmaf(w[1], z1, fmaf(w[2], z2, b[0])));
    float t1 = fmaf(w[3], z0, fmaf(w[4], z1, fmaf(w[5], z2, b[1])));
    float t2 = fmaf(w[6], z0, fmaf(w[7], z1, fmaf(w[8], z2, b[2])));
    // affine coupling from LDS-staged data
    float s0 = fast_sigmoid_p2(a[i * 6 + 0]);
    float s1 = fast_sigmoid_p2(a[i * 6 + 1]);
    float s2 = fast_sigmoid_p2(a[i * 6 + 2]);
    z0 = fmaf(t0, s0, a[i * 6 + 3]);
    z1 = fmaf(t1, s1, a[i * 6 + 4]);
    z2 = fmaf(t2, s2, a[i * 6 + 5]);
    // log s0 + log s1 + log s2 == log(s0*s1*s2): one v_log instead of three
    total += __logf(s0 * s1 * s2);
  }

  // final linear layer + standard-normal log density (constant pre-folded)
  {
    const float* w = Ws + kLayers * 9;
    const float* b = bs + kLayers * 3;
    float f0 = fmaf(w[0], z0, fmaf(w[1], z1, fmaf(w[2], z2, b[0])));
    float f1 = fmaf(w[3], z0, fmaf(w[4], z1, fmaf(w[5], z2, b[1])));
    float f2 = fmaf(w[6], z0, fmaf(w[7], z1, fmaf(w[8], z2, b[2])));
    total = fmaf(-0.5f, fmaf(f0, f0, fmaf(f1, f1, f2 * f2)), total);

    __builtin_nontemporal_store(f0, &outZ[row * 3 + 0]);
    __builtin_nontemporal_store(f1, &outZ[row * 3 + 1]);
    __builtin_nontemporal_store(f2, &outZ[row * 3 + 2]);
    __builtin_nontemporal_store(total, &outT[row]);
  }
}

extern "C" void kernel_launch(void* const* d_in, const int* in_sizes, int n_in,
                              void* d_out, int out_size, void* d_ws, size_t ws_size,
                              hipStream_t stream) {
  const float* x   = (const float*)d_in[0];   // (N, 3)
  const float* aff = (const float*)d_in[1];   // (N, 60)
  const float* Ws  = (const float*)d_in[2];   // (11, 3, 3)
  const float* bs  = (const float*)d_in[3];   // (11, 3)

  const int n = in_sizes[0] / 3;
  float* outZ = (float*)d_out;                    // first N*3 floats
  float* outT = (float*)d_out + (size_t)n * 3;    // then N floats
  float* cdet = (float*)d_ws;                     // 4 B of scratch

  logdet_prepass<<<1, 32, 0, stream>>>(Ws, cdet);

  const int grid = (n + kBlock - 1) / kBlock;
  flow_fwd_kernel<<<grid, kBlock, 0, stream>>>(x, aff, Ws, bs, cdet, outZ, outT, n);
}